// Quaduplet2Loss_11562051961166
// MI455X (gfx1250) — compile-verified
//
#include <hip/hip_runtime.h>
#include <hip/hip_bf16.h>

#define B_ROWS 8192
#define DIM 256
#define MARGINF 0.3f
#define BIGV 1e9f
#define SENT_THR 1e8f       // real squared distances are <= ~4.1
#define NCHUNK 8            // j chunks (grid.y), 1024 columns each
#define ROWS_PER_BLOCK 64   // i rows per block (4 waves x 16 rows)
#define WAVES_PER_BLOCK 4
#define TILES_PER_CHUNK 64  // 64 tiles * 16 cols = 1024 cols per chunk
#define ASTRIDE 258         // padded LDS row stride (floats) -> no bank conflicts

typedef __attribute__((ext_vector_type(2))) float v2f;
typedef __attribute__((ext_vector_type(8))) float v8f;

// ---------------------------------------------------------------------------
// Kernel 1: row-wise L2 normalize; also store sq[i] = sum(x_hat^2)
// ---------------------------------------------------------------------------
__global__ __launch_bounds__(256) void normalize_kernel(
    const float* __restrict__ in, float* __restrict__ xn, float* __restrict__ sq) {
  __shared__ float red[256];
  const int row = blockIdx.x;
  const int tid = threadIdx.x;
  const float v = in[(size_t)row * DIM + tid];
  red[tid] = v * v;
  __syncthreads();
  for (int s = 128; s > 0; s >>= 1) {
    if (tid < s) red[tid] += red[tid + s];
    __syncthreads();
  }
  const float ssum = red[0];
  const float norm = fmaxf(sqrtf(ssum), 1e-12f);
  xn[(size_t)row * DIM + tid] = v / norm;
  if (tid == 0) sq[row] = ssum / (norm * norm);
}

// ---------------------------------------------------------------------------
// Kernel 2: fused Gram (WMMA fp32 16x16x4) + masked min/max mining on SQUARED
// distances (sqrt is monotonic -> applied once per row in finalize).
// Branchless selects keep EXEC all-ones (required around WMMA) and avoid the
// saveexec/branch serialization the previous version compiled to.
// Partial row results (d^2 domain, +-BIGV sentinels) -> parts[4][NCHUNK][B].
// ---------------------------------------------------------------------------
__global__ __launch_bounds__(WAVES_PER_BLOCK * 32) void gram_mine_kernel(
    const float* __restrict__ xn, const float* __restrict__ sq,
    const int* __restrict__ tgt, float* __restrict__ parts) {
  __shared__ float As[ROWS_PER_BLOCK * ASTRIDE];

  const int tid  = threadIdx.x;
  const int lane = tid & 31;
  const int wv   = tid >> 5;   // wave id 0..3
  const int h    = lane >> 4;  // half-wave (0/1)
  const int n    = lane & 15;  // column / M index within half
  const int i0   = blockIdx.x * ROWS_PER_BLOCK;
  const int chunk = blockIdx.y;

  // Stage this wave's 16 A rows into LDS (padded stride).
  for (int e = lane; e < 16 * (DIM / 4); e += 32) {
    const int r  = e / (DIM / 4);
    const int c4 = (e % (DIM / 4)) * 4;
    const float4 src = *reinterpret_cast<const float4*>(
        &xn[(size_t)(i0 + wv * 16 + r) * DIM + c4]);
    float* dst = &As[(wv * 16 + r) * ASTRIDE + c4];
    dst[0] = src.x; dst[1] = src.y; dst[2] = src.z; dst[3] = src.w;
  }
  __syncthreads();

  // i-side per-row constants for the C layout rows: M = v + 8*h
  float sqi[8];
  int   ti[8];
#pragma unroll
  for (int v = 0; v < 8; ++v) {
    const int irow = i0 + wv * 16 + v + 8 * h;
    sqi[v] = sq[irow];
    ti[v]  = tgt[irow];
  }

  float ap[8], an[8], ap2[8], an2[8];
#pragma unroll
  for (int v = 0; v < 8; ++v) {
    ap[v] = -BIGV; an[v] = BIGV; ap2[v] = -BIGV; an2[v] = BIGV;
  }

  // A fragment base: lane holds A[M = n][K = k0 + 2h + {0,1}]
  const int aBase = (wv * 16 + n) * ASTRIDE + 2 * h;

  for (int t = 0; t < TILES_PER_CHUNK; ++t) {
    const int j0 = chunk * (TILES_PER_CHUNK * 16) + t * 16;
    const int j  = j0 + n;
    const float sqj = sq[j];
    const int   tj  = tgt[j];
    // B fragment: lane holds B[K = k0 + 2h + {0,1}][N = n] = xn[j][k...]
    const float* brow = &xn[(size_t)j * DIM + 2 * h];

    if (j0 + 16 < B_ROWS)  // uniform branch: prefetch next tile's rows
      __builtin_prefetch(&xn[(size_t)(j0 + 16 + n) * DIM], 0, 1);

    v8f acc = {};
#pragma unroll 8
    for (int k0 = 0; k0 < DIM; k0 += 4) {
      const float2 afv = *reinterpret_cast<const float2*>(&As[aBase + k0]);
      const float2 bfv = *reinterpret_cast<const float2*>(&brow[k0]);
      v2f a; a.x = afv.x; a.y = afv.y;
      v2f b; b.x = bfv.x; b.y = bfv.y;
      acc = __builtin_amdgcn_wmma_f32_16x16x4_f32(
          false, a, false, b, (short)0, acc, false, false);
    }

    // Branchless fused epilogue on squared distances.
    const bool vj = (tj >= 0);
#pragma unroll
    for (int v = 0; v < 8; ++v) {
      const float d2   = sqi[v] + sqj - 2.0f * acc[v];
      const bool vij   = (ti[v] >= 0) && vj;
      const bool same  = vij && (ti[v] == tj);
      const bool diff  = vij && (ti[v] != tj);
      ap[v]  = fmaxf(ap[v],  same ? d2 : -BIGV);
      an[v]  = fminf(an[v],  diff ? d2 :  BIGV);
      ap2[v] = fmaxf(ap2[v], vj   ? d2 : -BIGV);
      an2[v] = fminf(an2[v], vj   ?  BIGV : d2);
    }
  }

  // Butterfly reduce over the 16 lanes of each half-wave (masks < 16 never
  // cross the half boundary in wave32).
#pragma unroll
  for (int v = 0; v < 8; ++v) {
#pragma unroll
    for (int m = 8; m >= 1; m >>= 1) {
      ap[v]  = fmaxf(ap[v],  __shfl_xor(ap[v],  m, 32));
      an[v]  = fminf(an[v],  __shfl_xor(an[v],  m, 32));
      ap2[v] = fmaxf(ap2[v], __shfl_xor(ap2[v], m, 32));
      an2[v] = fminf(an2[v], __shfl_xor(an2[v], m, 32));
    }
  }

  if (n == 0) {  // lanes 0 and 16 write rows v and v+8
#pragma unroll
    for (int v = 0; v < 8; ++v) {
      const int row = i0 + wv * 16 + v + 8 * h;
      parts[(size_t)0 * NCHUNK * B_ROWS + chunk * B_ROWS + row] = ap[v];
      parts[(size_t)1 * NCHUNK * B_ROWS + chunk * B_ROWS + row] = an[v];
      parts[(size_t)2 * NCHUNK * B_ROWS + chunk * B_ROWS + row] = ap2[v];
      parts[(size_t)3 * NCHUNK * B_ROWS + chunk * B_ROWS + row] = an2[v];
    }
  }
}

// ---------------------------------------------------------------------------
// Kernel 3: combine chunk partials (d^2 domain), convert to distances once
// per row (sentinel-preserving), apply margin loss, write scalar.
// ---------------------------------------------------------------------------
__global__ __launch_bounds__(256) void finalize_kernel(
    const int* __restrict__ tgt, const float* __restrict__ parts,
    float* __restrict__ out) {
  __shared__ float s0[256], s1[256], s2[256], s3[256];
  const int tid = threadIdx.x;
  float si = 0.f, sb = 0.f, nv = 0.f, nb = 0.f;
  for (int r = tid; r < B_ROWS; r += 256) {
    float apd2 = -BIGV, and2 = BIGV, ap2d2 = -BIGV, an2d2 = BIGV;
    for (int ch = 0; ch < NCHUNK; ++ch) {
      apd2  = fmaxf(apd2,  parts[(size_t)0 * NCHUNK * B_ROWS + ch * B_ROWS + r]);
      and2  = fminf(and2,  parts[(size_t)1 * NCHUNK * B_ROWS + ch * B_ROWS + r]);
      ap2d2 = fmaxf(ap2d2, parts[(size_t)2 * NCHUNK * B_ROWS + ch * B_ROWS + r]);
      an2d2 = fminf(an2d2, parts[(size_t)3 * NCHUNK * B_ROWS + ch * B_ROWS + r]);
    }
    // sqrt(clip(d2, 1e-12)) with +-BIG sentinel preservation (monotonic map)
    const float ap  = (apd2  > -SENT_THR) ? sqrtf(fmaxf(apd2,  1e-12f)) : -BIGV;
    const float an  = (and2  <  SENT_THR) ? sqrtf(fmaxf(and2,  1e-12f)) :  BIGV;
    const float ap2 = (ap2d2 > -SENT_THR) ? sqrtf(fmaxf(ap2d2, 1e-12f)) : -BIGV;
    const float an2 = (an2d2 <  SENT_THR) ? sqrtf(fmaxf(an2d2, 1e-12f)) :  BIGV;

    const int t = tgt[r];
    if (t >= 0) {
      si += fmaxf(ap  - an  + MARGINF, 0.f);
      sb += fmaxf(ap2 - an2 + MARGINF, 0.f);
      nv += 1.f;
    } else {
      nb += 1.f;
    }
  }
  s0[tid] = si; s1[tid] = sb; s2[tid] = nv; s3[tid] = nb;
  __syncthreads();
  for (int s = 128; s > 0; s >>= 1) {
    if (tid < s) {
      s0[tid] += s0[tid + s];
      s1[tid] += s1[tid + s];
      s2[tid] += s2[tid + s];
      s3[tid] += s3[tid + s];
    }
    __syncthreads();
  }
  if (tid == 0) {
    const float denom = fmaxf(s2[0], 1.f);
    float loss = s0[0] / denom;                        // INSTANCE_WEIGHT = 1.0
    if (s3[0] > 0.f) loss += 0.25f * (s1[0] / denom);  // BG_WEIGHT gated on n_bg>0
    out[0] = (s2[0] > 0.f) ? loss : 0.f;
  }
}

// ---------------------------------------------------------------------------
// Launch: ws = [xn: 8192*256 f32][sq: 8192 f32][parts: 4*8*8192 f32] ~9.5 MB
// ---------------------------------------------------------------------------
extern "C" void kernel_launch(void* const* d_in, const int* in_sizes, int n_in,
                              void* d_out, int out_size, void* d_ws, size_t ws_size,
                              hipStream_t stream) {
  (void)in_sizes; (void)n_in; (void)out_size; (void)ws_size;
  const float* inputs  = (const float*)d_in[0];
  const int*   targets = (const int*)d_in[1];
  float* out = (float*)d_out;

  float* xn    = (float*)d_ws;
  float* sqv   = xn + (size_t)B_ROWS * DIM;
  float* parts = sqv + B_ROWS;

  normalize_kernel<<<B_ROWS, 256, 0, stream>>>(inputs, xn, sqv);
  gram_mine_kernel<<<dim3(B_ROWS / ROWS_PER_BLOCK, NCHUNK),
                     WAVES_PER_BLOCK * 32, 0, stream>>>(xn, sqv, targets, parts);
  finalize_kernel<<<1, 256, 0, stream>>>(targets, parts, out);
}